// NondominatedPartitioning_65584150610421
// MI455X (gfx1250) — compile-verified
//
#include <hip/hip_runtime.h>
#include <stdint.h>

// ---------------- configuration ----------------
#define B_BUCKETS 65536          // y1 value-range buckets (256KB of u32 keys)
#define TILE_PTS  4096           // points (float2) per TDM tile -> 32KB LDS per buffer
#define TILE_ELEMS (TILE_PTS * 2)
#define BLK       256            // 8 waves (wave32)
#define G_STREAM  512            // blocks for the two streaming passes (~4.8 tiles each)
#define G_HV      256            // blocks for the pairwise survivor kernel
#define MAXC      (1u << 20)     // survivor capacity cap (8 MB)

typedef unsigned int uint32x4 __attribute__((ext_vector_type(4)));
typedef int          int32x4  __attribute__((ext_vector_type(4)));
typedef int          int32x8  __attribute__((ext_vector_type(8)));

// order-preserving float -> uint key (for exact atomicMin in key space)
__device__ __forceinline__ uint32_t fkey(float f) {
  uint32_t b = __float_as_uint(f);
  return (b & 0x80000000u) ? ~b : (b | 0x80000000u);
}

// monotone (nondecreasing) bucket map, clamped; k(a) < k(b)  =>  a < b
__device__ __forceinline__ int bucket_of(float y1) {
  float t = floorf((y1 + 16.0f) * (float)(B_BUCKETS / 32));
  t = fminf(fmaxf(t, 0.0f), (float)(B_BUCKETS - 1));
  return (int)t;
}

// ---------------- Tensor Data Mover: 1-row tile of TILE_ELEMS f32, global->LDS ----
// D# per CDNA5 ISA ch.8: group0 = {flags, lds_addr, global_addr, type=2},
// group1 = {data_size=4B, tensor_dim0=remaining elems (tail OOB -> zero, never
// consumed), tensor_dim1=1, tile_dim0=TILE_ELEMS, tile_dim1=1, stride0=TILE_ELEMS}.
__device__ __forceinline__ void tdm_load_tile(const float* gsrc,
                                              uint32_t lds_byte_off,
                                              uint32_t rem_elems) {
  uint64_t ga = (uint64_t)(uintptr_t)gsrc;
  uint32x4 g0;
  g0.x = 0x1u;                                  // count=1, is_restore=0, gather off
  g0.y = lds_byte_off;                          // lds_addr (bytes)
  g0.z = (uint32_t)ga;                          // global_addr[31:0]
  g0.w = ((uint32_t)(ga >> 32) & 0x01FFFFFFu)   // global_addr[56:32]
         | 0x80000000u;                         // type=2 ("image") in bits 127:126
  int32x8 g1;
  g1[0] = 0x00020000;                                    // data_size=2 -> 4 bytes
  g1[1] = (int)((rem_elems & 0xFFFFu) << 16);            // tensor_dim0[15:0]
  g1[2] = (int)((rem_elems >> 16) | (1u << 16));         // dim0[31:16] | tensor_dim1=1
  g1[3] = (int)((uint32_t)TILE_ELEMS << 16);             // dim1 hi=0 | tile_dim0
  g1[4] = 1;                                             // tile_dim1=1, tile_dim2=0
  g1[5] = TILE_ELEMS;                                    // tensor_dim0_stride lo
  g1[6] = 0;
  g1[7] = 0;                                             // tensor_dim1_stride
  int32x4 z4 = {0, 0, 0, 0};
#if __has_include(<hip/amd_detail/amd_gfx1250_TDM.h>)
  int32x8 z8 = {0, 0, 0, 0, 0, 0, 0, 0};
  __builtin_amdgcn_tensor_load_to_lds(g0, g1, z4, z4, z8, 0);
#else
  __builtin_amdgcn_tensor_load_to_lds(g0, g1, z4, z4, 0);
#endif
}

// ---------------- stage 0: init workspace ----------------
__global__ void k_init(uint32_t* bucketMin, uint32_t* counter, double* partials) {
  uint32_t i = blockIdx.x * blockDim.x + threadIdx.x;
  uint32_t stride = gridDim.x * blockDim.x;
  for (uint32_t j = i; j < B_BUCKETS; j += stride) bucketMin[j] = 0xFFFFFFFFu;
  for (uint32_t j = i; j < G_HV; j += stride) partials[j] = 0.0;
  if (i == 0) *counter = 0u;
}

// ---------------- stage 1: per-bucket min of y0 key (TDM double-buffered stream) ----
__global__ void k_bucketmin(const float* __restrict__ Y, uint32_t N,
                            uint32_t* __restrict__ bucketMin) {
  __shared__ float bufA[TILE_ELEMS];
  __shared__ float bufB[TILE_ELEMS];
  const bool isW0 = (threadIdx.x < 32);
  const uint32_t ldsA = (uint32_t)(uintptr_t)&bufA[0];
  const uint32_t ldsB = (uint32_t)(uintptr_t)&bufB[0];
  const uint32_t T = (N + TILE_PTS - 1) / TILE_PTS;
  uint32_t t = blockIdx.x;
  if (t >= T) return;
  if (isW0) tdm_load_tile(Y + (size_t)t * TILE_ELEMS, ldsA, 2u * (N - t * TILE_PTS));
  uint32_t cur = 0;
  for (; t < T; t += gridDim.x) {
    uint32_t tn = t + gridDim.x;
    if (isW0) {
      if (tn < T) {   // prefetch next tile into the other buffer, wait for current
        tdm_load_tile(Y + (size_t)tn * TILE_ELEMS, cur ? ldsA : ldsB,
                      2u * (N - tn * TILE_PTS));
        __builtin_amdgcn_s_wait_tensorcnt((short)1);
      } else {
        __builtin_amdgcn_s_wait_tensorcnt((short)0);
      }
    }
    __syncthreads();
    const float* buf = cur ? bufB : bufA;
    const uint32_t valid = min((uint32_t)TILE_PTS, N - t * TILE_PTS);
    for (uint32_t j = threadIdx.x; j < valid; j += blockDim.x) {
      float y0 = -buf[2 * j + 0];
      float y1 = -buf[2 * j + 1];
      atomicMin(&bucketMin[bucket_of(y1)], fkey(y0));
    }
    __syncthreads();
    cur ^= 1u;
  }
}

// ---------------- stage 2: exclusive prefix-min over buckets (1 block) ----------
__global__ void k_scan(const uint32_t* __restrict__ bucketMin,
                       uint32_t* __restrict__ Mexcl) {
  __shared__ uint32_t s[1024];
  __shared__ uint32_t carry_s;
  if (threadIdx.x == 0) carry_s = 0xFFFFFFFFu;
  __syncthreads();
  for (int base = 0; base < B_BUCKETS; base += 1024) {
    s[threadIdx.x] = bucketMin[base + threadIdx.x];
    __syncthreads();
    for (int off = 1; off < 1024; off <<= 1) {          // Hillis-Steele min-scan
      uint32_t u = (threadIdx.x >= (unsigned)off) ? s[threadIdx.x - off] : 0xFFFFFFFFu;
      __syncthreads();
      s[threadIdx.x] = min(s[threadIdx.x], u);
      __syncthreads();
    }
    uint32_t carry = carry_s;
    uint32_t excl = (threadIdx.x == 0) ? carry : min(carry, s[threadIdx.x - 1]);
    Mexcl[base + threadIdx.x] = excl;
    __syncthreads();
    if (threadIdx.x == 1023) carry_s = min(carry, s[1023]);
    __syncthreads();
  }
}

// ---------------- stage 3: filter dominated points, compact survivors ------------
__global__ void k_filter(const float* __restrict__ Y, uint32_t N,
                         const uint32_t* __restrict__ Mexcl,
                         uint32_t* __restrict__ counter,
                         float2* __restrict__ surv, uint32_t cap) {
  __shared__ float bufA[TILE_ELEMS];
  __shared__ float bufB[TILE_ELEMS];
  const bool isW0 = (threadIdx.x < 32);
  const uint32_t ldsA = (uint32_t)(uintptr_t)&bufA[0];
  const uint32_t ldsB = (uint32_t)(uintptr_t)&bufB[0];
  const uint32_t T = (N + TILE_PTS - 1) / TILE_PTS;
  uint32_t t = blockIdx.x;
  if (t >= T) return;
  if (isW0) tdm_load_tile(Y + (size_t)t * TILE_ELEMS, ldsA, 2u * (N - t * TILE_PTS));
  uint32_t cur = 0;
  for (; t < T; t += gridDim.x) {
    uint32_t tn = t + gridDim.x;
    if (isW0) {
      if (tn < T) {
        tdm_load_tile(Y + (size_t)tn * TILE_ELEMS, cur ? ldsA : ldsB,
                      2u * (N - tn * TILE_PTS));
        __builtin_amdgcn_s_wait_tensorcnt((short)1);
      } else {
        __builtin_amdgcn_s_wait_tensorcnt((short)0);
      }
    }
    __syncthreads();
    const float* buf = cur ? bufB : bufA;
    const uint32_t valid = min((uint32_t)TILE_PTS, N - t * TILE_PTS);
    for (uint32_t j = threadIdx.x; j < valid; j += blockDim.x) {
      float y0 = -buf[2 * j + 0];
      float y1 = -buf[2 * j + 1];
      // keep unless some strictly-smaller-y1 bucket already beats y0 strictly
      if (fkey(y0) <= Mexcl[bucket_of(y1)]) {
        uint32_t pos = atomicAdd(counter, 1u);
        if (pos < cap) surv[pos] = make_float2(y0, y1);
      }
    }
    __syncthreads();
    cur ^= 1u;
  }
}

// ---------------- stage 4: exact sweep on survivors (O(S^2), S ~ thousands) -------
__global__ void k_hv(const float2* __restrict__ surv,
                     const uint32_t* __restrict__ counter, uint32_t cap,
                     const float* __restrict__ ref, double* __restrict__ partials) {
  __shared__ float2 tile[2048];
  __shared__ double red[BLK];
  const uint32_t S = min(*counter, cap);
  const float r0 = -ref[0];
  const float r1 = -ref[1];
  double acc = 0.0;
  const uint32_t stride = gridDim.x * blockDim.x;
  for (uint32_t base = blockIdx.x * blockDim.x; base < S; base += stride) {
    const uint32_t i = base + threadIdx.x;
    const bool act = (i < S);
    float y0 = 0.f, y1 = 0.f;
    if (act) { y0 = surv[i].x; y1 = surv[i].y; }
    float L = r1;  // running min of y1 over lexicographic predecessors
    for (uint32_t jb = 0; jb < S; jb += 2048) {
      const uint32_t cnt = min(2048u, S - jb);
      __syncthreads();
      for (uint32_t j = threadIdx.x; j < cnt; j += blockDim.x) tile[j] = surv[jb + j];
      __syncthreads();
      if (act) {
        for (uint32_t j = 0; j < cnt; ++j) {
          const float a0 = tile[j].x, a1 = tile[j].y;
          const uint32_t gj = jb + j;
          const bool prec = (a0 < y0) ||
                            (a0 == y0 && (a1 < y1 || (a1 == y1 && gj < i)));
          if (prec) L = fminf(L, a1);
        }
      }
    }
    if (act) {
      const float w = fmaxf(r0 - y0, 0.0f);
      const float h = fmaxf(L - y1, 0.0f);
      acc += (double)w * (double)h;
    }
  }
  red[threadIdx.x] = acc;
  __syncthreads();
  for (int off = BLK / 2; off > 0; off >>= 1) {
    if ((int)threadIdx.x < off) red[threadIdx.x] += red[threadIdx.x + off];
    __syncthreads();
  }
  if (threadIdx.x == 0) partials[blockIdx.x] = red[0];
}

// ---------------- stage 5: fixed-order final sum -> f32 ----------------
__global__ void k_final(const double* __restrict__ partials, int n,
                        float* __restrict__ out) {
  if (blockIdx.x == 0 && threadIdx.x == 0) {
    double s = 0.0;
    for (int i = 0; i < n; ++i) s += partials[i];
    out[0] = (float)s;
  }
}

extern "C" void kernel_launch(void* const* d_in, const int* in_sizes, int n_in,
                              void* d_out, int out_size, void* d_ws, size_t ws_size,
                              hipStream_t stream) {
  const float* Y   = (const float*)d_in[0];
  const float* ref = (const float*)d_in[1];
  const uint32_t N = (uint32_t)(in_sizes[0] / 2);

  char* ws = (char*)d_ws;
  const size_t off_bmin = 0;
  const size_t off_mex  = (size_t)B_BUCKETS * 4;          // 256 KB
  const size_t off_cnt  = (size_t)B_BUCKETS * 8;          // 512 KB
  const size_t off_part = off_cnt + 256;
  const size_t off_surv = off_part + (size_t)G_HV * 8 + 256;

  uint32_t* bucketMin = (uint32_t*)(ws + off_bmin);
  uint32_t* Mexcl     = (uint32_t*)(ws + off_mex);
  uint32_t* counter   = (uint32_t*)(ws + off_cnt);
  double*   partials  = (double*)(ws + off_part);
  float2*   surv      = (float2*)(ws + off_surv);

  uint32_t cap = 0;
  if (ws_size > off_surv + sizeof(float2)) {
    size_t c = (ws_size - off_surv) / sizeof(float2);
    cap = (uint32_t)(c < (size_t)MAXC ? c : (size_t)MAXC);
  }

  k_init<<<256, 256, 0, stream>>>(bucketMin, counter, partials);
  k_bucketmin<<<G_STREAM, BLK, 0, stream>>>(Y, N, bucketMin);
  k_scan<<<1, 1024, 0, stream>>>(bucketMin, Mexcl);
  k_filter<<<G_STREAM, BLK, 0, stream>>>(Y, N, Mexcl, counter, surv, cap);
  k_hv<<<G_HV, BLK, 0, stream>>>(surv, counter, cap, ref, partials);
  k_final<<<1, 32, 0, stream>>>(partials, G_HV, (float*)d_out);
}